// HashEncoding_53618371723964
// MI455X (gfx1250) — compile-verified
//
#include <hip/hip_runtime.h>
#include <math.h>

#define NLVL 16
#define HASH_MASK ((1u << 19) - 1u)

struct LevelCfg { unsigned off; float scale; unsigned res; unsigned hashed; };
struct HashCfg  { LevelCfg lv[NLVL]; };

__global__ __launch_bounds__(256) void hashenc_kernel(
    const float* __restrict__ x,
    const float* __restrict__ grid,
    float* __restrict__ out,
    int npoints,
    HashCfg cfg)
{
    // Stage per-level config through LDS once; hot path reads ds_load_b128,
    // avoiding divergent-indexed kernarg access in the loop.
    __shared__ LevelCfg sL[NLVL];
    if (threadIdx.x < NLVL) sL[threadIdx.x] = cfg.lv[threadIdx.x];
    __syncthreads();

    const int tid = blockIdx.x * blockDim.x + threadIdx.x;
    const int lvl = tid & (NLVL - 1);            // 16 lanes = 16 levels of one point
    const LevelCfg L = sL[lvl];

    const unsigned off  = L.off;
    const float    scl  = L.scale;
    const unsigned res  = L.res;
    const unsigned res2 = res * res;
    const bool     isH  = (L.hashed != 0);

    const int pstride = (int)((gridDim.x * blockDim.x) >> 4);
    const float2* __restrict__ g2 = (const float2*)grid;

    for (int p = tid >> 4; p < npoints; p += pstride) {
        // Prefetch next iteration's point coords (global_prefetch_b8).
        const int pn = p + pstride;
        if (pn < npoints)
            __builtin_prefetch(x + (size_t)pn * 3, 0, 0);

        // Streaming input: non-temporal loads (don't pollute L2; the 48.8MB
        // feature table must stay L2-resident on the 192MB L2).
        const float* xp = x + (size_t)p * 3;
        const float x0 = __builtin_nontemporal_load(xp + 0);
        const float x1 = __builtin_nontemporal_load(xp + 1);
        const float x2 = __builtin_nontemporal_load(xp + 2);

        const float p0 = scl * x0 + 0.5f;
        const float p1 = scl * x1 + 0.5f;
        const float p2 = scl * x2 + 0.5f;
        const float f0 = floorf(p0), f1 = floorf(p1), f2 = floorf(p2);
        const float fr0 = p0 - f0, fr1 = p1 - f1, fr2 = p2 - f2;
        const unsigned b0 = (unsigned)f0, b1 = (unsigned)f1, b2 = (unsigned)f2;

        const float wx1 = fr0, wx0 = 1.0f - fr0;
        const float wy1 = fr1, wy0 = 1.0f - fr1;
        const float wz1 = fr2, wz0 = 1.0f - fr2;

        float acc0 = 0.0f, acc1 = 0.0f;
#pragma unroll
        for (int k = 0; k < 8; ++k) {
            // corner order matches product([0,1],repeat=3): dim2 (z) fastest
            const unsigned bx = (k >> 2) & 1u;
            const unsigned by = (k >> 1) & 1u;
            const unsigned bz =  k       & 1u;
            const unsigned c0 = b0 + bx;
            const unsigned c1 = b1 + by;
            const unsigned c2 = b2 + bz;
            const float w = (bx ? wx1 : wx0) * (by ? wy1 : wy0) * (bz ? wz1 : wz0);

            const unsigned idxh = (c0 ^ (c1 * 2654435761u) ^ (c2 * 805459861u)) & HASH_MASK;
            const unsigned idxd = c0 + c1 * res + c2 * res2;
            const unsigned idx  = isH ? idxh : idxd;   // branchless (v_cndmask)

            const float2 v = g2[(size_t)off + idx];    // global_load_b64, RT hint: L2-resident
            acc0 = fmaf(v.x, w, acc0);
            acc1 = fmaf(v.y, w, acc1);
        }

        // 16 lanes of a half-wave write one contiguous 128B line.
        // Non-temporal: 256MB output stream must not evict the table from L2.
        float2 r; r.x = acc0; r.y = acc1;
        const double bits = __builtin_bit_cast(double, r);
        __builtin_nontemporal_store(bits, (double*)(out + (size_t)p * 32 + lvl * 2));
    }
}

extern "C" void kernel_launch(void* const* d_in, const int* in_sizes, int n_in,
                              void* d_out, int out_size, void* d_ws, size_t ws_size,
                              hipStream_t stream) {
    const float* x    = (const float*)d_in[0];
    const float* grid = (const float*)d_in[1];
    float*       out  = (float*)d_out;
    const int npoints = in_sizes[0] / 3;

    // Rebuild level config each call with the same double-precision libm
    // sequence as the Python reference (deterministic, no caching).
    HashCfg cfg;
    {
        const double per = exp(log(2048.0 * 1.0 / 16.0) / 15.0);
        const double l2b = log2(per);
        long long n = 0;
        const long long hsz = 1ll << 19;
        for (int i = 0; i < NLVL; ++i) {
            const double scale = pow(2.0, (double)i * l2b) * 16.0 - 1.0;
            const int resi = (int)ceil(scale) + 1;
            const double r3 = pow((double)resi, 3.0);
            long long p = (long long)(ceil(r3 / 8.0) * 8.0);
            if (p > hsz) p = hsz;
            cfg.lv[i].off    = (unsigned)n;
            cfg.lv[i].scale  = (float)scale;
            cfg.lv[i].res    = (unsigned)resi;
            cfg.lv[i].hashed = (r3 > (double)p) ? 1u : 0u;
            n += p;
        }
    }

    const int threads = 256;                   // 8 waves (wave32) per block
    long long total = (long long)npoints * NLVL;
    int blocks = (int)((total + threads - 1) / threads);
    if (blocks > 8192) blocks = 8192;          // grid-stride: enables prefetch + cfg hoist
    if (blocks < 1) blocks = 1;

    hipLaunchKernelGGL(hashenc_kernel, dim3(blocks), dim3(threads), 0, stream,
                       x, grid, out, npoints, cfg);
}